// AttentionKernelBlock_45088566673546
// MI455X (gfx1250) — compile-verified
//
#include <hip/hip_runtime.h>
#include <hip/hip_bf16.h>

typedef __attribute__((ext_vector_type(2))) float v2f;
typedef __attribute__((ext_vector_type(8))) float v8f;

#define BATCH   2
#define CH      64
#define DDIM    48
#define HDIM    96
#define WDIM    96
#define SPATIAL (DDIM * HDIM * WDIM)   /* 442368, divisible by 16 */
#define NSHIFT  27

__device__ __forceinline__ v8f wmma_f32_step(v2f a, v2f b, v8f c) {
  // V_WMMA_F32_16X16X4_F32 : D(16x16 f32) = A(16x4 f32) * B(4x16 f32) + C
  return __builtin_amdgcn_wmma_f32_16x16x4_f32(
      /*neg_a=*/false, a, /*neg_b=*/false, b,
      /*c_mod=*/(short)0, c, /*reuse_a=*/false, /*reuse_b=*/false);
}

// ---------------------------------------------------------------------------
// K1: qkv projection  QKV[192,S] = qkv_w[192,64] * X[64,S]  (+bias)
// One wave computes one 16(M)x16(N) tile via 16 WMMA k-steps (K=64).
// grid = (S/16/8, 12, BATCH), block = 256 (8 waves).
// ---------------------------------------------------------------------------
__global__ void __launch_bounds__(256)
qkv_gemm_kernel(const float* __restrict__ x, const float* __restrict__ w,
                const float* __restrict__ bias,
                float* __restrict__ q, float* __restrict__ k,
                float* __restrict__ v) {
  const int lane  = threadIdx.x & 31;
  const int wave  = threadIdx.x >> 5;
  const int hi    = lane >> 4;     // lane group (0: lanes 0-15, 1: lanes 16-31)
  const int ln    = lane & 15;
  const int nbase = (blockIdx.x * 8 + wave) * 16;
  const int mbase = blockIdx.y * 16;
  const int b     = blockIdx.z;
  const float* xb = x + (size_t)b * CH * SPATIAL;
  const int m     = mbase + ln;    // A-matrix row handled by this lane

  v8f acc = {};
#pragma unroll
  for (int kk = 0; kk < 16; ++kk) {
    const int k0 = kk * 4 + 2 * hi;
    v2f a;                                   // A 16x4: vgpr0 K=k0, vgpr1 K=k0+1
    a.x = w[m * CH + k0];
    a.y = w[m * CH + k0 + 1];
    v2f bb;                                  // B 4x16: rows k0,k0+1 across lanes
    bb.x = xb[(size_t)k0 * SPATIAL + nbase + ln];
    bb.y = xb[(size_t)(k0 + 1) * SPATIAL + nbase + ln];
    acc = wmma_f32_step(a, bb, acc);
  }

  float* dst;
  int o0;
  if (mbase < 64)       { dst = q; o0 = mbase; }
  else if (mbase < 128) { dst = k; o0 = mbase - 64; }
  else                  { dst = v; o0 = mbase - 128; }

#pragma unroll
  for (int r = 0; r < 8; ++r) {
    const int row = r + 8 * hi;              // C/D: vgpr r -> rows r, r+8
    dst[(size_t)b * CH * SPATIAL + (size_t)(o0 + row) * SPATIAL + nbase + ln] =
        acc[r] + bias[mbase + row];
  }
}

// ---------------------------------------------------------------------------
// K2: local 3x3x3 attention, one thread per voxel.
// logits_s = sum_c Q[c]*(Kpad_s[c] + rel[c,s]); rel contributes even in the
// zero-padded region (matches reference). y = x + sum_s softmax_s * Vpad_s.
// y is written over the Q buffer (each thread only touches its own column).
// ---------------------------------------------------------------------------
__global__ void __launch_bounds__(256)
attn_kernel(const float* __restrict__ x, const float* __restrict__ qws,
            const float* __restrict__ kws, const float* __restrict__ vws,
            const float* __restrict__ rel, float* __restrict__ yws) {
  const long long tid = (long long)blockIdx.x * blockDim.x + threadIdx.x;
  const int b  = (int)(tid / SPATIAL);
  const int s  = (int)(tid % SPATIAL);
  const int d  = s / (HDIM * WDIM);
  const int rm = s % (HDIM * WDIM);
  const int h  = rm / WDIM;
  const int wc = rm % WDIM;
  const size_t base = (size_t)b * CH * SPATIAL + s;

  int offs[NSHIFT];
  unsigned valid = 0;
  {
    int idx = 0;
#pragma unroll
    for (int di = -1; di <= 1; ++di)
#pragma unroll
      for (int dj = -1; dj <= 1; ++dj)
#pragma unroll
        for (int dk = -1; dk <= 1; ++dk) {
          offs[idx] = di * (HDIM * WDIM) + dj * WDIM + dk;
          const bool ok = ((unsigned)(d + di) < DDIM) &&
                          ((unsigned)(h + dj) < HDIM) &&
                          ((unsigned)(wc + dk) < WDIM);
          valid |= (ok ? 1u : 0u) << idx;
          ++idx;
        }
  }

  float logits[NSHIFT];
#pragma unroll
  for (int i = 0; i < NSHIFT; ++i) logits[i] = 0.f;

  for (int c = 0; c < CH; ++c) {
    const float  qc   = qws[base + (size_t)c * SPATIAL];
    const float* kc   = kws + base + (size_t)c * SPATIAL;
    const float* relc = rel + c * NSHIFT;
    __builtin_prefetch(kc + SPATIAL, 0, 0);  // global_prefetch for next plane
#pragma unroll
    for (int si = 0; si < NSHIFT; ++si) {
      const float kv = ((valid >> si) & 1u) ? kc[offs[si]] : 0.f;
      logits[si] = fmaf(qc, kv + relc[si], logits[si]);
    }
  }

  const float scale = 0.125f;  // 1/sqrt(64)
  float mx = -3.402823466e38f;
#pragma unroll
  for (int si = 0; si < NSHIFT; ++si) {
    logits[si] *= scale;
    mx = fmaxf(mx, logits[si]);
  }
  float sum = 0.f;
#pragma unroll
  for (int si = 0; si < NSHIFT; ++si) {
    const float e = __expf(logits[si] - mx);
    logits[si] = e;
    sum += e;
  }
  const float inv = 1.0f / sum;
#pragma unroll
  for (int si = 0; si < NSHIFT; ++si) logits[si] *= inv;

  for (int c = 0; c < CH; ++c) {
    float acc = x[base + (size_t)c * SPATIAL];   // residual
    const float* vc = vws + base + (size_t)c * SPATIAL;
    __builtin_prefetch(vc + SPATIAL, 0, 0);
#pragma unroll
    for (int si = 0; si < NSHIFT; ++si)
      if ((valid >> si) & 1u) acc = fmaf(logits[si], vc[offs[si]], acc);
    yws[base + (size_t)c * SPATIAL] = acc;
  }
}

// ---------------------------------------------------------------------------
// K3: out projection  OUT[64,S] = out_w[64,64] * Y[64,S] (+bias)
// grid = (S/16/8, 4, BATCH), block = 256.
// ---------------------------------------------------------------------------
__global__ void __launch_bounds__(256)
out_gemm_kernel(const float* __restrict__ y, const float* __restrict__ w,
                const float* __restrict__ bias, float* __restrict__ out) {
  const int lane  = threadIdx.x & 31;
  const int wave  = threadIdx.x >> 5;
  const int hi    = lane >> 4;
  const int ln    = lane & 15;
  const int nbase = (blockIdx.x * 8 + wave) * 16;
  const int mbase = blockIdx.y * 16;
  const int b     = blockIdx.z;
  const float* yb = y + (size_t)b * CH * SPATIAL;
  const int m     = mbase + ln;

  v8f acc = {};
#pragma unroll
  for (int kk = 0; kk < 16; ++kk) {
    const int k0 = kk * 4 + 2 * hi;
    v2f a;
    a.x = w[m * CH + k0];
    a.y = w[m * CH + k0 + 1];
    v2f bb;
    bb.x = yb[(size_t)k0 * SPATIAL + nbase + ln];
    bb.y = yb[(size_t)(k0 + 1) * SPATIAL + nbase + ln];
    acc = wmma_f32_step(a, bb, acc);
  }

#pragma unroll
  for (int r = 0; r < 8; ++r) {
    const int row = r + 8 * hi;
    out[(size_t)b * CH * SPATIAL + (size_t)(mbase + row) * SPATIAL + nbase + ln] =
        acc[r] + bias[mbase + row];
  }
}

// ---------------------------------------------------------------------------
extern "C" void kernel_launch(void* const* d_in, const int* in_sizes, int n_in,
                              void* d_out, int out_size, void* d_ws, size_t ws_size,
                              hipStream_t stream) {
  const float* x     = (const float*)d_in[0];
  const float* qkv_w = (const float*)d_in[1];
  const float* qkv_b = (const float*)d_in[2];
  const float* out_w = (const float*)d_in[3];
  const float* out_b = (const float*)d_in[4];
  const float* rel   = (const float*)d_in[5];

  // Workspace: [Q | K | V], each BATCH*CH*SPATIAL floats (~226.5 MB).
  // y aliases Q (Q fully consumed per-column before overwrite in K2).
  float* ws = (float*)d_ws;
  const size_t plane = (size_t)BATCH * CH * SPATIAL;
  float* qws = ws;
  float* kws = ws + plane;
  float* vws = ws + 2 * plane;
  float* yws = qws;

  dim3 blk(256);

  {  // qkv projection (WMMA f32): 12 M-tiles of 16 over 192 out-channels
    dim3 grd(SPATIAL / 16 / 8, 12, BATCH);
    qkv_gemm_kernel<<<grd, blk, 0, stream>>>(x, qkv_w, qkv_b, qws, kws, vws);
  }
  {  // fused attention (softmax over 27 shifts) + residual, writes y
    dim3 grd((unsigned)((size_t)BATCH * SPATIAL / 256));
    attn_kernel<<<grd, blk, 0, stream>>>(x, qws, kws, vws, rel, yws);
  }
  {  // out projection (WMMA f32): 4 M-tiles of 16 over 64 out-channels
    dim3 grd(SPATIAL / 16 / 8, 4, BATCH);
    out_gemm_kernel<<<grd, blk, 0, stream>>>(yws, out_w, out_b, (float*)d_out);
  }
}